// OptimizedSpectralPredictor_2216203124850
// MI455X (gfx1250) — compile-verified
//
#include <hip/hip_runtime.h>

typedef __attribute__((ext_vector_type(2))) float v2f;
typedef __attribute__((ext_vector_type(8))) float v8f;

#define Zn  128
#define Yn  512
#define Xn  512
#define Pn  15
#define NWt 19            // 4 spatial + 15 spectral weights per band
#define YX  (Yn * Xn)

// Band-matrix coefficient for WMMA A tile of z-block z0:
//   output row m (z = z0+m), input column k (z_in = z0-16+k), lag d = m+16-k
__device__ __forceinline__ float wcoef(const float* __restrict__ w,
                                       int z0, int m, int k) {
  int d    = m + 16 - k;
  int z_in = z0 - 16 + k;
  if (d >= 1 && d <= Pn && z_in >= 0)
    return w[(z0 + m) * NWt + (NWt - d)];
  return 0.0f;
}

__global__ __launch_bounds__(256) void spectral_pred_wmma(
    const float* __restrict__ img,   // (128, 512, 512)
    const float* __restrict__ wts,   // (128, 19)
    float* __restrict__ pred_out,    // (128, 512, 512)
    float* __restrict__ res_out) {   // (128, 512, 512)
  // Per-z-block band matrices, stored in exact WMMA A-register layout
  __shared__ v2f sA[8][8][32];                 // [zblock][kchunk][lane]  16 KB
  __shared__ __align__(16) float sW4[Zn][4];   // spatial weights          2 KB

  const int tid = threadIdx.x;

  // ---- Stage 1: build A matrices (2048 v2f entries; 8 per thread) ----
  for (int t = 0; t < 8; ++t) {
    int e    = tid + t * 256;        // 0..2047
    int lane = e & 31;
    int j    = (e >> 5) & 7;         // k-chunk
    int zb   = e >> 8;               // z-block
    int z0   = zb * 16;
    int m    = lane & 15;
    int kb   = 4 * j + ((lane >> 4) << 1);
    v2f a;
    a.x = wcoef(wts, z0, m, kb);
    a.y = wcoef(wts, z0, m, kb + 1);
    sA[zb][j][lane] = a;
  }
  {
    int f0 = tid * 2;                // 512 spatial-weight floats total
    sW4[f0 >> 2][f0 & 3] = wts[(f0 >> 2) * NWt + (f0 & 3)];
    int f1 = f0 + 1;
    sW4[f1 >> 2][f1 & 3] = wts[(f1 >> 2) * NWt + (f1 & 3)];
  }
  __syncthreads();

  // ---- Stage 2: each wave owns one 16-pixel x-strip, loops all z-blocks ----
  const int wave    = tid >> 5;
  const int lane    = tid & 31;
  const int strip   = blockIdx.x * 8 + wave;   // 16384 strips total
  const int px_base = strip * 16;              // strips never cross a row
  const int px      = px_base + (lane & 15);   // this lane's pixel column (N)
  const int x       = px & (Xn - 1);
  const int y       = px >> 9;
  const int m_off   = (lane >> 4) * 8;         // C-layout: rows m..m+7 (+8 hi half)
  const int kb_off  = (lane >> 4) << 1;        // A/B k striping across lane halves

  // Branchless boundary masks (only z==0 rows can go truly out of bounds;
  // offsets are clamped to >=0 there and the value forced to 0 via cndmask).
  const float mask_n  = (y > 0) ? 1.0f : 0.0f;
  const float mask_w  = (x > 0) ? 1.0f : 0.0f;
  const float mask_nw = mask_n * mask_w;

  // B register file with cross-iteration rotation: chunks 0..3 of block zb
  // equal chunks 4..7 of block zb-1 (z ranges overlap by 16 bands).
  // For zb==0, chunks 0..3 cover z_in in [-16,-1] -> zeros (weights there are
  // zero too, but keep B finite).
  v2f b[8];
#pragma unroll
  for (int j = 0; j < 4; ++j) { b[j].x = 0.0f; b[j].y = 0.0f; }

  for (int zb = 0; zb < 8; ++zb) {
    const int z0 = zb * 16;

    // A tiles from LDS (ds_load_b64 pairs, already in WMMA register layout)
    v2f a[8];
#pragma unroll
    for (int j = 0; j < 8; ++j) a[j] = sA[zb][j][lane];

    // Load only the new upper half of B: chunks 4..7 -> z_in in [z0, z0+15]
#pragma unroll
    for (int j = 0; j < 4; ++j) {
      int zi = z0 + 4 * j + kb_off;          // always >= 0
      b[j + 4].x = img[zi * YX + px];
      b[j + 4].y = img[(zi + 1) * YX + px];
    }

    // Spectral banded matmul: C(16x16) += W(16x32) x Img(32x16), f32 exact
    v8f c = {};
#pragma unroll
    for (int j = 0; j < 8; ++j) {
      c = __builtin_amdgcn_wmma_f32_16x16x4_f32(
          /*neg_a=*/false, a[j], /*neg_b=*/false, b[j],
          /*c_mod=*/(short)0, c, /*reuse_a=*/false, /*reuse_b=*/false);
    }

    // Spatial terms + clip + stores; C layout: VGPR v -> (M = m_off+v, N = px)
#pragma unroll
    for (int v = 0; v < 8; ++v) {
      int   z    = z0 + m_off + v;
      int   base = z * YX + px;
      // clamp offsets (only z==0,y==0 edge can underflow) -> no exec branches
      int off_n  = base - Xn;     off_n  = off_n  < 0 ? 0 : off_n;
      int off_w  = base - 1;      off_w  = off_w  < 0 ? 0 : off_w;
      int off_nw = base - Xn - 1; off_nw = off_nw < 0 ? 0 : off_nw;
      float self = img[base];
      float n    = img[off_n]  * mask_n;
      float w    = img[off_w]  * mask_w;
      float nw   = img[off_nw] * mask_nw;
      float4 w4  = *reinterpret_cast<const float4*>(&sW4[z][0]);
      float fourth = (n + w + nw) * (1.0f / 3.0f);
      float p = c[v] + n * w4.x + w * w4.y + nw * w4.z + fourth * w4.w;
      p = fminf(fmaxf(p, -32768.0f), 32767.0f);   // clip to 16-bit dynamic range
      pred_out[base] = p;
      res_out[base]  = self - p;
    }

    // Rotate B for the next z-block (overlapping 16 input bands)
#pragma unroll
    for (int j = 0; j < 4; ++j) b[j] = b[j + 4];
  }
}

extern "C" void kernel_launch(void* const* d_in, const int* in_sizes, int n_in,
                              void* d_out, int out_size, void* d_ws, size_t ws_size,
                              hipStream_t stream) {
  const float* img = (const float*)d_in[0];   // (128,512,512) f32
  const float* wts = (const float*)d_in[1];   // (128,19) f32
  float* pred = (float*)d_out;                // outputs concatenated: pred, residuals
  float* res  = pred + (size_t)Zn * YX;

  // 16384 strips / 8 waves per block -> 2048 blocks of 256 threads (8 wave32)
  hipLaunchKernelGGL(spectral_pred_wmma, dim3(2048), dim3(256), 0, stream,
                     img, wts, pred, res);
}